// RMFA_90366111908219
// MI455X (gfx1250) — compile-verified
//
#include <hip/hip_runtime.h>

// Problem constants (from reference)
#define BB 2
#define HH 8
#define LL 2048
#define DD 64
#define PP 128
#define PS 32            // P-slice width
#define NP 4             // number of P slices (PP/PS)
#define CK 16            // chunk length along L
#define NCH (LL / CK)
#define ROWP (DD + 4)    // padded tile row (68 floats = 272B, 16B aligned)
#define NORM 0.35355339059327379f   // 64^(-0.25)
#define EPSV 1.0f

typedef __attribute__((ext_vector_type(2))) float v2f;
typedef __attribute__((ext_vector_type(8))) float v8f;
typedef __attribute__((ext_vector_type(4))) int   v4i;

#if defined(__gfx1250__) &&                                        \
    __has_builtin(__builtin_amdgcn_global_load_async_to_lds_b128) && \
    __has_builtin(__builtin_amdgcn_s_wait_asynccnt)
#define USE_ASYNC 1
#else
#define USE_ASYNC 0
#endif

__device__ __forceinline__ v8f wmma4(v2f a, v2f b, v8f c) {
  // D = A(16x4, f32) * B(4x16, f32) + C(16x16, f32)
  return __builtin_amdgcn_wmma_f32_16x16x4_f32(false, a, false, b, (short)0, c,
                                               false, false);
}

#if USE_ASYNC
typedef __attribute__((address_space(1))) v4i gv4i;   // global int4
typedef __attribute__((address_space(3))) v4i sv4i;   // LDS int4
__device__ __forceinline__ void async_cp16(const float* g, float* l) {
  __builtin_amdgcn_global_load_async_to_lds_b128((gv4i*)g, (sv4i*)l, 0, 0);
}
#endif

// One workgroup per (b,h,ps). 4 waves (128 threads); wave w owns output
// columns d = 16w..16w+15 and the matching 32x16 slab of state S (two v8f
// WMMA accumulators held in registers for the entire L loop). Q/K/V chunk
// tiles are double-buffered in LDS and prefetched with async copies.
__global__ __launch_bounds__(128)
void rfa_main(const float* __restrict__ Q, const float* __restrict__ K,
              const float* __restrict__ V, const float* __restrict__ W,
              const int* __restrict__ mask,
              float* __restrict__ out_part, float* __restrict__ den_part)
{
  __shared__ float qkv[2][3][CK][ROWP];  // double-buffered Q,K,V chunk tiles
  __shared__ float Ws[PS][DD + 1];       // W slice [32][64] (pre-scaled)
  __shared__ float pqs[CK][PS + 1];      // projected q chunk [16][32]
  __shared__ float pks[CK][PS + 1];      // projected (masked) k chunk
  __shared__ float Ss[PS][DD + 1];       // LDS mirror of state (B-operand form)
  __shared__ float Ams[CK][CK + 1];      // causal-masked scores
  __shared__ float cums[CK][PS + 1];     // z + inclusive prefix of pk
  __shared__ float zs[PS];
  __shared__ float dens[CK];

  const int bid  = blockIdx.x;
  const int ps   = bid % NP;
  const int bh   = bid / NP;
  const int b    = bh / HH;
  const int tid  = threadIdx.x;
  const int wid  = tid >> 5;           // wave id 0..3
  const int lane = tid & 31;
  const int nl   = lane & 15;          // N / M index within tile
  const int hi   = lane >> 4;          // lane-half selects K pair / M+8

  const float* Qp = Q + (size_t)bh * LL * DD;
  const float* Kp = K + (size_t)bh * LL * DD;
  const float* Vp = V + (size_t)bh * LL * DD;
  const int*   mp = mask + (size_t)b * LL;
  float* Op = out_part + ((size_t)ps * (BB * HH) + bh) * LL * DD;
  float* Dp = den_part + ((size_t)ps * (BB * HH) + bh) * LL;

  // Stage W slice (fold in the d^-0.25 normalizer; projection is linear so
  // this applies it to both pq and pk); zero state mirror and z.
  for (int e = tid; e < PS * DD; e += 128) {
    int p = e >> 6, d = e & 63;
    Ws[p][d] = W[(size_t)(ps * PS + p) * DD + d] * NORM;
    Ss[p][d] = 0.0f;
  }
  if (tid < PS) zs[tid] = 0.0f;

#if USE_ASYNC
  // Issue async prefetch of one chunk's Q/K/V into LDS buffer slot s.
  auto prefetch = [&](int cc, int s) {
    for (int it = 0; it < 2; ++it) {
      int e  = tid + it * 128;        // vec4 index 0..255 within 16x64 tile
      int r  = e >> 4;
      int c4 = (e & 15) << 2;
      size_t goff = (size_t)(cc * CK + r) * DD + c4;
      async_cp16(Qp + goff, &qkv[s][0][r][c4]);
      async_cp16(Kp + goff, &qkv[s][1][r][c4]);
      async_cp16(Vp + goff, &qkv[s][2][r][c4]);
    }
  };
  prefetch(0, 0);
#endif

  const int dcol = wid * 16;
  v8f s0 = {};   // state rows p=0..15, cols dcol..dcol+15
  v8f s1 = {};   // state rows p=16..31

  for (int ch = 0; ch < NCH; ++ch) {
    const int l0  = ch * CK;
    const int cur = ch & 1;
    float (*Qc)[ROWP] = qkv[cur][0];
    float (*Kc)[ROWP] = qkv[cur][1];
    float (*Vc)[ROWP] = qkv[cur][2];

#if USE_ASYNC
    // Prefetch next chunk into the other buffer, then wait until this
    // chunk's 6 async copies (issued last iteration) have landed.
    if (ch + 1 < NCH) {
      prefetch(ch + 1, cur ^ 1);
      __builtin_amdgcn_s_wait_asynccnt(6);
    } else {
      __builtin_amdgcn_s_wait_asynccnt(0);
    }
#else
    for (int e = tid; e < CK * DD; e += 128) {
      int r = e >> 6, c = e & 63;
      Qc[r][c] = Qp[(size_t)(l0 + r) * DD + c];
      Kc[r][c] = Kp[(size_t)(l0 + r) * DD + c];
      Vc[r][c] = Vp[(size_t)(l0 + r) * DD + c];
    }
#endif
    __syncthreads();

    // ---- projection: waves 0,1 -> pq tiles; waves 2,3 -> pk tiles ----
    {
      const float (*src)[ROWP] = (wid < 2) ? Qc : Kc;
      const int n0 = (wid & 1) * 16;
      v8f acc = {};
      for (int k0 = 0; k0 < DD; k0 += 4) {
        v2f a, bm;
        a.x  = src[nl][k0 + 2 * hi];            // A[m][k], m = nl
        a.y  = src[nl][k0 + 2 * hi + 1];
        bm.x = Ws[n0 + nl][k0 + 2 * hi];        // B[k=d][n=p] = W[p][d]
        bm.y = Ws[n0 + nl][k0 + 2 * hi + 1];
        acc = wmma4(a, bm, acc);
      }
      if (wid < 2) {
        for (int r = 0; r < 8; ++r)
          pqs[r + 8 * hi][n0 + nl] = acc[r];
      } else {
        for (int r = 0; r < 8; ++r) {           // mask keys (linear, so
          int i = r + 8 * hi;                   // post-projection is exact)
          pks[i][n0 + nl] = (mp[l0 + i] != 0) ? acc[r] : 0.0f;
        }
      }
    }
    __syncthreads();

    // ---- phase A: inter-chunk readout O += pq @ S_prev (all waves) ----
    v8f o = {};
    for (int k0 = 0; k0 < PS; k0 += 4) {
      v2f a, bm;
      a.x  = pqs[nl][k0 + 2 * hi];
      a.y  = pqs[nl][k0 + 2 * hi + 1];
      bm.x = Ss[k0 + 2 * hi][dcol + nl];        // B[k=p][n=d]
      bm.y = Ss[k0 + 2 * hi + 1][dcol + nl];
      o = wmma4(a, bm, o);
    }
    if (wid == 0) {
      // intra-chunk scores A = pq @ pk^T, causal mask, share via LDS
      v8f asc = {};
      for (int k0 = 0; k0 < PS; k0 += 4) {
        v2f a, bm;
        a.x  = pqs[nl][k0 + 2 * hi];
        a.y  = pqs[nl][k0 + 2 * hi + 1];
        bm.x = pks[nl][k0 + 2 * hi];            // B[k=p][n=j] = pk[j][p]
        bm.y = pks[nl][k0 + 2 * hi + 1];
        asc = wmma4(a, bm, asc);
      }
      for (int r = 0; r < 8; ++r) {
        int i = r + 8 * hi;
        Ams[i][nl] = (nl <= i) ? asc[r] : 0.0f; // keep j <= i
      }
    } else if (wid == 1) {
      // cums[i][p] = z[p] + sum_{j<=i} pk[j][p]; update z; zero dens
      float running = zs[lane];
      for (int i = 0; i < CK; ++i) {
        running += pks[i][lane];
        cums[i][lane] = running;
      }
      zs[lane] = running;
      if (lane < CK) dens[lane] = 0.0f;
    }
    __syncthreads();

    // ---- phase B: intra-chunk O += Am @ V ----
    for (int k0 = 0; k0 < CK; k0 += 4) {
      v2f a, bm;
      a.x  = Ams[nl][k0 + 2 * hi];
      a.y  = Ams[nl][k0 + 2 * hi + 1];
      bm.x = Vc[k0 + 2 * hi][dcol + nl];
      bm.y = Vc[k0 + 2 * hi + 1][dcol + nl];
      o = wmma4(a, bm, o);
    }
    // ---- state update: S += pk^T @ V (A[m=p][k=c] = pk[c][p]) ----
    for (int k0 = 0; k0 < CK; k0 += 4) {
      v2f a0, a1, bm;
      a0.x = pks[k0 + 2 * hi][nl];
      a0.y = pks[k0 + 2 * hi + 1][nl];
      a1.x = pks[k0 + 2 * hi][16 + nl];
      a1.y = pks[k0 + 2 * hi + 1][16 + nl];
      bm.x = Vc[k0 + 2 * hi][dcol + nl];
      bm.y = Vc[k0 + 2 * hi + 1][dcol + nl];
      s0 = wmma4(a0, bm, s0);
      s1 = wmma4(a1, bm, s1);
    }
    // mirror updated state to LDS for next chunk's phase A
    for (int r = 0; r < 8; ++r) {
      Ss[r + 8 * hi][dcol + nl]      = s0[r];
      Ss[16 + r + 8 * hi][dcol + nl] = s1[r];
    }
    // denominator partials: den[i] = sum_p pq[i][p] * cums[i][p]
    {
      int i  = tid >> 3;
      int p0 = (tid & 7) * 4;
      float part = 0.0f;
      for (int pp = 0; pp < 4; ++pp)
        part += pqs[i][p0 + pp] * cums[i][p0 + pp];
      atomicAdd(&dens[i], part);
    }
    __syncthreads();

    // ---- write this chunk's partials to global ----
    for (int r = 0; r < 8; ++r) {
      int i = r + 8 * hi;
      Op[(size_t)(l0 + i) * DD + dcol + nl] = o[r];
    }
    if (wid == 1 && lane < CK) Dp[l0 + lane] = dens[lane];
  }
}

// Reduce the NP feature-slice partials, apply max(den, EPS) and divide.
__global__ __launch_bounds__(256)
void rfa_reduce(const float* __restrict__ out_part,
                const float* __restrict__ den_part,
                float* __restrict__ out)
{
  const size_t total = (size_t)BB * HH * LL * DD;
  size_t e = (size_t)blockIdx.x * 256 + threadIdx.x;
  if (e >= total) return;
  size_t bhl = e / DD;
  const size_t strideO = (size_t)BB * HH * LL * DD;
  const size_t strideD = (size_t)BB * HH * LL;
  float num = 0.0f, den = 0.0f;
  for (int p = 0; p < NP; ++p) {
    num += out_part[(size_t)p * strideO + e];
    den += den_part[(size_t)p * strideD + bhl];
  }
  out[e] = num / fmaxf(den, EPSV);
}

extern "C" void kernel_launch(void* const* d_in, const int* in_sizes, int n_in,
                              void* d_out, int out_size, void* d_ws, size_t ws_size,
                              hipStream_t stream)
{
  const float* Q    = (const float*)d_in[0];
  const float* K    = (const float*)d_in[1];
  const float* V    = (const float*)d_in[2];
  const float* W    = (const float*)d_in[3];
  const int*   mask = (const int*)d_in[4];
  float* out = (float*)d_out;

  // Workspace: numerator partials [NP][B,H,L,D] then denominator partials
  // [NP][B,H,L]  (~34 MB total).
  float* out_part = (float*)d_ws;
  float* den_part = out_part + (size_t)NP * BB * HH * LL * DD;

  dim3 g1(BB * HH * NP), b1(128);
  rfa_main<<<g1, b1, 0, stream>>>(Q, K, V, W, mask, out_part, den_part);

  const size_t total = (size_t)BB * HH * LL * DD;
  dim3 g2((unsigned)((total + 255) / 256)), b2(256);
  rfa_reduce<<<g2, b2, 0, stream>>>(out_part, den_part, out);
}